// Heston_46909632807650
// MI455X (gfx1250) — compile-verified
//
#include <hip/hip_runtime.h>
#include <math.h>

// ---------------------------------------------------------------------------
// Heston window log-likelihood, MI455X (gfx1250, wave32).
//   kprep    : parallel — dlog[i] = log(sp[i+1])-log(sp[i]); block partials of
//              (sp[i+1]-sp[i])/sp[i] for mu (LDS tree reduction).
//   kscan    : 1 wave — fixed-order sum of partials -> mu; lane 0 runs the
//              inherently serial variance recurrence (latency-minimized log2-
//              domain chain) with double-buffered float4 dlog loads +
//              global_prefetch; writes var path.
//   klogpdf  : parallel — logpdf terms; wave reduction via V_WMMA_F32_16X16X4_F32
//              (B = ones => cross-lane sums), inter-wave combine through LDS.
//   kfinal   : 1 wave — fixed-order combine of block partials -> d_out[0].
// ---------------------------------------------------------------------------

#define WINDOW   262144
#define NB1      1024              // WINDOW / 256
#define NB3      32
#define DT       (1.0f / 252.0f)
#define TWO_PI_F 6.28318530717958647692f

#define LOG2E_F    1.4426950408889634f
#define SQRT_LN2_F 0.8325546111576977f
#define RSQRT_DT_F 15.874507866387544f   // sqrt(252)

typedef __attribute__((ext_vector_type(2))) float v2f;
typedef __attribute__((ext_vector_type(8))) float v8f;

struct HParams { float ekdt, c1, c2, c3, Crho, halfrho2; };

__device__ __forceinline__ HParams make_params(const float* l_k, const float* l_theta,
                                               const float* l_sigma, const float* at_rho) {
    HParams p;
    float k     = __expf(*l_k);
    float theta = __expf(*l_theta);
    float sigma = __expf(*l_sigma);
    float rho   = tanhf(*at_rho);
    p.ekdt      = __expf(-k * DT);
    float om    = 1.0f - p.ekdt;
    float s2    = sigma * sigma;
    p.c1        = s2 * p.ekdt * om / k;
    p.c2        = theta * s2 * om * om / (2.0f * k);
    p.c3        = theta * om;                         // m = fma(v_p, ekdt, c3)
    p.Crho      = rho * (LOG2E_F * SQRT_LN2_F * RSQRT_DT_F);
    p.halfrho2  = 0.5f * rho * rho;
    return p;
}

// One step of the serial variance recurrence, entirely in log2 domain.
// Dependent chain: max -> fma -> mul -> rcp -> fma -> log2 -> sqrt -> fma
//                  -> exp2 -> cndmask.  (compare t2>1 runs parallel to exp2;
//                  rsq(v_p), b, log2(m) are off-chain branches)
//   raw = m*exp(E) = exp2(t2),  t2 = sqrt(u)*Crho*rsq(v_p)*b - halfrho2*u + log2(m)
//   with u = log2(1 + s2/m^2)   (ln2*log2e == 1 collapses the rho^2 term)
//   guard raw>2  <=>  t2>1.
__device__ __forceinline__ float heston_step(float vr, float dl, const HParams& P,
                                             float mu_dt) {
    float v_p = fmaxf(vr, 0.01f);                               // == relu(vr-.01)+.01
    float m   = fmaf(v_p, P.ekdt, P.c3);
    float s2  = fmaf(v_p, P.c1, P.c2);
    float q   = fmaf(s2, __builtin_amdgcn_rcpf(m * m), 1.0f);
    float u   = __builtin_amdgcn_logf(q);                       // log2(q), q>1 => u>0
    float st  = __builtin_amdgcn_sqrtf(u);
    float ab  = (P.Crho * __builtin_amdgcn_rsqf(v_p))           // off-chain branch
                * fmaf(0.5f * DT, v_p, dl - mu_dt);
    float inner = fmaf(-P.halfrho2, u, __builtin_amdgcn_logf(m));
    float t2  = fmaf(st, ab, inner);
    float raw = __builtin_amdgcn_exp2f(t2);
    return (t2 > 1.0f) ? vr : raw;
}

// Wave32 sum reduction through the matrix unit:
//   A(16x4) = per-lane partials (VGPR0=x, VGPR1=0), B(4x16) = ones
//   => D[m][n] = x[m] + x[m+16].  Summing the 8 D VGPRs per lane and one
//   shfl_xor(16) yields the full 32-lane sum in every lane.
__device__ __forceinline__ float wave_reduce_wmma(float x) {
    v2f a; a.x = x;    a.y = 0.0f;
    v2f b; b.x = 1.0f; b.y = 1.0f;
    v8f c = {};
    c = __builtin_amdgcn_wmma_f32_16x16x4_f32(false, a, false, b,
                                              (short)0, c, false, false);
    float s = c[0] + c[1] + c[2] + c[3] + c[4] + c[5] + c[6] + c[7];
    s += __shfl_xor(s, 16, 32);
    return s;
}

// ---------------------------------------------------------------------------
__global__ void kprep(const float* __restrict__ sp,
                      float* __restrict__ dlog,
                      float* __restrict__ part) {
    __shared__ float sdata[256];
    int i = blockIdx.x * 256 + threadIdx.x;          // i in [0, WINDOW)
    float s0 = sp[i];
    float s1 = sp[i + 1];
    float r  = (s1 - s0) / s0;
    if (i < WINDOW - 1) dlog[i] = __logf(s1) - __logf(s0);
    sdata[threadIdx.x] = r;
    __syncthreads();
    #pragma unroll
    for (int off = 128; off > 0; off >>= 1) {
        if (threadIdx.x < off) sdata[threadIdx.x] += sdata[threadIdx.x + off];
        __syncthreads();
    }
    if (threadIdx.x == 0) part[blockIdx.x] = sdata[0];
}

// ---------------------------------------------------------------------------
__global__ void kscan(const float* __restrict__ part,
                      const float* l_k, const float* l_theta,
                      const float* l_sigma, const float* at_rho,
                      const float* v0p,
                      const float* __restrict__ dlog,
                      float* __restrict__ var,
                      float* __restrict__ mu_out) {
    int lane = threadIdx.x;                          // 32 threads, 1 wave
    float s = 0.0f;
    for (int j = lane; j < NB1; j += 32) s += part[j];
    #pragma unroll
    for (int off = 16; off > 0; off >>= 1) s += __shfl_xor(s, off, 32);
    float mu = s * (1.0f / (float)WINDOW);
    if (lane != 0) return;

    mu_out[0] = mu;
    HParams P   = make_params(l_k, l_theta, l_sigma, at_rho);
    float mu_dt = mu * DT;

    float vr = v0p[0];
    var[0] = vr;

    const float4* dl4 = (const float4*)dlog;
    const int nq = (WINDOW - 1) / 4;                 // 65535 full quads
    float4 cur = dl4[0];
    int idx = 1;
    for (int q = 0; q < nq; ++q) {
        float4 nxt = dl4[q + 1];                     // in-bounds: dlog has WINDOW floats
        __builtin_prefetch(&dl4[q + 16], 0, 3);      // global_prefetch ahead
        vr = heston_step(vr, cur.x, P, mu_dt); var[idx]     = vr;
        vr = heston_step(vr, cur.y, P, mu_dt); var[idx + 1] = vr;
        vr = heston_step(vr, cur.z, P, mu_dt); var[idx + 2] = vr;
        vr = heston_step(vr, cur.w, P, mu_dt); var[idx + 3] = vr;
        idx += 4;
        cur = nxt;
    }
    for (int i = nq * 4; i < WINDOW - 1; ++i) {      // 3-step tail
        vr = heston_step(vr, dlog[i], P, mu_dt);
        var[idx++] = vr;
    }
}

// ---------------------------------------------------------------------------
__global__ void klogpdf(const float* __restrict__ sp,
                        const float* __restrict__ var,
                        const float* __restrict__ mu_p,
                        float* __restrict__ part3) {
    __shared__ float swave[8];
    int tid  = threadIdx.x;
    int gid0 = blockIdx.x * 256 + tid;
    float mu = mu_p[0];
    float acc = 0.0f;
    for (int i = gid0; i < WINDOW; i += 256 * NB3) { // uniform trips: EXEC stays all-ones
        float v    = var[i];
        float d    = __logf(sp[i + 1]) - __logf(sp[i]);
        float vi   = v * DT;
        float diff = d - (mu - 0.5f * v) * DT;
        float lp   = -0.5f * diff * diff / vi - 0.5f * __logf(TWO_PI_F * vi);
        float tr   = fmaxf(__expf(lp) - 1e-6f, 0.0f) + 1e-6f;
        acc += __logf(tr);
    }
    float wsum = wave_reduce_wmma(acc);
    int wave = tid >> 5;
    if ((tid & 31) == 0) swave[wave] = wsum;
    __syncthreads();
    if (tid == 0) {
        float b = 0.0f;
        #pragma unroll
        for (int w = 0; w < 8; ++w) b += swave[w];   // fixed order -> deterministic
        part3[blockIdx.x] = b;
    }
}

// ---------------------------------------------------------------------------
__global__ void kfinal(const float* __restrict__ part3, float* __restrict__ out) {
    int lane = threadIdx.x;                          // 32 threads == NB3
    float s = part3[lane];
    #pragma unroll
    for (int off = 16; off > 0; off >>= 1) s += __shfl_xor(s, off, 32);
    if (lane == 0) out[0] = s;
}

// ---------------------------------------------------------------------------
// Fallback: fully serial single-thread version if d_ws is too small.
__global__ void kmono(const float* __restrict__ sp,
                      const float* l_k, const float* l_theta,
                      const float* l_sigma, const float* at_rho,
                      const float* v0p, float* __restrict__ out) {
    float musum = 0.0f;
    for (int i = 0; i < WINDOW; ++i) musum += (sp[i + 1] - sp[i]) / sp[i];
    float mu = musum * (1.0f / (float)WINDOW);

    HParams P   = make_params(l_k, l_theta, l_sigma, at_rho);
    float mu_dt = mu * DT;

    float vr  = v0p[0];
    float acc = 0.0f;
    float lprev = __logf(sp[0]);
    for (int i = 0; i < WINDOW; ++i) {
        float lnext = __logf(sp[i + 1]);
        float d     = lnext - lprev;
        float v     = vr;
        float vi    = v * DT;
        float diff  = d - (mu - 0.5f * v) * DT;
        float lp    = -0.5f * diff * diff / vi - 0.5f * __logf(TWO_PI_F * vi);
        float tr    = fmaxf(__expf(lp) - 1e-6f, 0.0f) + 1e-6f;
        acc += __logf(tr);
        if (i < WINDOW - 1)
            vr = heston_step(vr, d, P, mu_dt);
        lprev = lnext;
    }
    out[0] = acc;
}

// ---------------------------------------------------------------------------
extern "C" void kernel_launch(void* const* d_in, const int* in_sizes, int n_in,
                              void* d_out, int out_size, void* d_ws, size_t ws_size,
                              hipStream_t stream) {
    const float* sp      = (const float*)d_in[0];
    const float* l_k     = (const float*)d_in[1];
    const float* l_theta = (const float*)d_in[2];
    const float* l_sigma = (const float*)d_in[3];
    const float* at_rho  = (const float*)d_in[4];
    const float* v0p     = (const float*)d_in[5];
    float*       out     = (float*)d_out;
    float*       ws      = (float*)d_ws;

    const size_t needF = (size_t)2 * WINDOW + NB1 + 1 + NB3 + 8;
    if (ws_size >= needF * sizeof(float)) {
        float* dlog  = ws;                            // WINDOW floats
        float* part  = ws + WINDOW;                   // NB1 floats
        float* var   = ws + WINDOW + NB1;             // WINDOW floats
        float* mu    = ws + (size_t)2 * WINDOW + NB1; // 1 float
        float* part3 = mu + 1;                        // NB3 floats

        kprep  <<<NB1, 256, 0, stream>>>(sp, dlog, part);
        kscan  <<<1,    32, 0, stream>>>(part, l_k, l_theta, l_sigma, at_rho, v0p,
                                         dlog, var, mu);
        klogpdf<<<NB3, 256, 0, stream>>>(sp, var, mu, part3);
        kfinal <<<1,    32, 0, stream>>>(part3, out);
    } else {
        kmono  <<<1,     1, 0, stream>>>(sp, l_k, l_theta, l_sigma, at_rho, v0p, out);
    }
}